// ROIAlign_81003083202760
// MI455X (gfx1250) — compile-verified
//
#include <hip/hip_runtime.h>

// ROI Align (single sample per bin): input (4,256,200,200) f32, 1024 ROIs,
// output (1024,256,7,7) f32. Memory-bound on MI455X (23.3 TB/s HBM, 192 MB L2
// >= 164 MB input). Strategy: keep input L2-resident (default RT loads),
// stream write-once output with non-temporal stores, and amortize all bin
// math over 4 channels per thread (channel stride folds into the 24-bit
// global-load immediate offset, so the extra channels cost only load+fma+store).

namespace {
constexpr int   kC      = 256;
constexpr int   kH      = 200;
constexpr int   kW      = 200;
constexpr int   kOutH   = 7;
constexpr int   kOutW   = 7;
constexpr float kScale  = 0.25f;
constexpr int   kHW     = kH * kW;               // 40000 floats (160000 B)
constexpr int   kCHW    = kC * kHW;              // 10,240,000
constexpr int   kBins   = kOutH * kOutW;         // 49
constexpr int   kPerRoi = kC * kBins;            // 12544
constexpr int   kKC     = 4;                     // channels per thread
constexpr int   kBlk    = 224;                   // 7 wave32 per block
constexpr int   kThreadsPerRoi = (kC / kKC) * kBins;  // 3136
constexpr int   kChunks = kThreadsPerRoi / kBlk;      // 14 blocks per ROI (exact)
} // namespace

__global__ __launch_bounds__(kBlk) void roi_align_f32_kernel(
    const float* __restrict__ inp,   // (4,256,200,200)
    const float* __restrict__ rois,  // (M,5) = b,x1,y1,x2,y2
    float* __restrict__ out)         // (M,256,7,7)
{
  // ---- block-uniform ROI parameters (scalar s_load + uniform math) ----
  const int m = (int)blockIdx.y;
  const float* __restrict__ r = rois + m * 5;
  const int   b  = (int)r[0];
  const float x1 = r[1] * kScale;
  const float y1 = r[2] * kScale;
  const float x2 = r[3] * kScale;
  const float y2 = r[4] * kScale;
  const float bin_w = fmaxf(x2 - x1, 1.0f) * (1.0f / (float)kOutW);
  const float bin_h = fmaxf(y2 - y1, 1.0f) * (1.0f / (float)kOutH);
  const float* __restrict__ img = inp + (size_t)b * kCHW;

  // ---- per-thread coordinate: j -> (channel group c4, bin s=(ph,pw)) ----
  const unsigned j  = (unsigned)blockIdx.x * kBlk + threadIdx.x;  // 0..3135
  const unsigned c4 = (j * 21400u) >> 20;        // j / 49, exact for j < 43690
  const unsigned s  = j - c4 * 49u;
  const unsigned ph = (s * 37u) >> 8;            // s / 7, exact for s < 86
  const unsigned pw = s - ph * 7u;

  // ---- sample point (matches reference math) ----
  const float ys = y1 + ((float)ph + 0.5f) * bin_h;
  const float xs = x1 + ((float)pw + 0.5f) * bin_w;
  const float py = ys * ((float)(kH - 1) / (float)kH);
  const float px = xs * ((float)(kW - 1) / (float)kW);
  const float fy = floorf(py);
  const float fx = floorf(px);
  const int   y0 = (int)fy;
  const int   x0 = (int)fx;
  const float ly = py - fy, lx = px - fx;
  const float hy = 1.0f - ly, hx = 1.0f - lx;

  // Validity folded into the weights: 0*v == reference's where(valid, v, 0).
  const bool okY0 = (y0 >= 0) & (y0 < kH);
  const bool okY1 = (y0 + 1 >= 0) & (y0 + 1 < kH);
  const bool okX0 = (x0 >= 0) & (x0 < kW);
  const bool okX1 = (x0 + 1 >= 0) & (x0 + 1 < kW);
  const float w00 = (okY0 & okX0) ? hy * hx : 0.0f;
  const float w01 = (okY0 & okX1) ? hy * lx : 0.0f;
  const float w10 = (okY1 & okX0) ? ly * hx : 0.0f;
  const float w11 = (okY1 & okX1) ? ly * lx : 0.0f;

  const int y0c = min(max(y0, 0), kH - 1);
  const int y1c = min(max(y0 + 1, 0), kH - 1);
  const int x0c = min(max(x0, 0), kW - 1);
  const int x1c = min(max(x0 + 1, 0), kW - 1);

  // 4 tap base pointers for channel c4*4; further channels use the 24-bit
  // immediate offset (k * 160000 B), costing no extra address math.
  const float* __restrict__ plane = img + (size_t)(c4 * kKC) * kHW;
  const float* __restrict__ row0  = plane + y0c * kW;
  const float* __restrict__ row1  = plane + y1c * kW;
  const float* __restrict__ p00   = row0 + x0c;
  const float* __restrict__ p01   = row0 + x1c;
  const float* __restrict__ p10   = row1 + x0c;
  const float* __restrict__ p11   = row1 + x1c;

  float* __restrict__ optr =
      out + (size_t)m * kPerRoi + (size_t)(c4 * kKC) * kBins + s;

#pragma unroll
  for (int k = 0; k < kKC; ++k) {
    const float v00 = p00[k * kHW];
    const float v01 = p01[k * kHW];
    const float v10 = p10[k * kHW];
    const float v11 = p11[k * kHW];
    const float res = w00 * v00 + w01 * v01 + w10 * v10 + w11 * v11;
    // Write-once output: non-temporal store keeps the input L2-resident.
    __builtin_nontemporal_store(res, optr + k * kBins);
  }
}

extern "C" void kernel_launch(void* const* d_in, const int* in_sizes, int n_in,
                              void* d_out, int out_size, void* d_ws, size_t ws_size,
                              hipStream_t stream) {
  (void)n_in; (void)out_size; (void)d_ws; (void)ws_size;
  const float* inp  = (const float*)d_in[0];
  const float* rois = (const float*)d_in[1];
  float*       out  = (float*)d_out;

  const int M = in_sizes[1] / 5;          // 1024 ROIs
  dim3 grid(kChunks, M);                  // 14 blocks x 224 threads x 4 ch = 1 ROI
  roi_align_f32_kernel<<<grid, kBlk, 0, stream>>>(inp, rois, out);
}